// AtomLayer_61177514164321
// MI455X (gfx1250) — compile-verified
//
#include <hip/hip_runtime.h>

#define AS1 __attribute__((address_space(1)))
#define AS3 __attribute__((address_space(3)))
typedef int v4i __attribute__((vector_size(16)));   // pointee type the builtins expect

constexpr int kB = 16, kS = 512, kN = 4096, kK = 64;
constexpr int kRows = kB * kS;              // 8192 rows
constexpr int kTPB  = 256;                  // 8 wave32 waves
constexpr int kVPT  = kN / kTPB;            // 16 elements per thread
constexpr long long kFeatElems = (long long)kRows * kN;  // 33,554,432
constexpr long long kIdxElems  = (long long)kRows * kK;  // 524,288

__global__ __launch_bounds__(kTPB) void topk_scatter_kernel(
    const float* __restrict__ x, const float* __restrict__ thr_p,
    float* __restrict__ out)
{
  __shared__ __align__(16) float buf[kN];   // row staging: x -> xm -> feat
  __shared__ float redV[kTPB / 32];
  __shared__ int   redI[kTPB / 32];
  __shared__ int   sBestI;

  const int row  = blockIdx.x;
  const int tid  = threadIdx.x;
  const int lane = tid & 31;
  const int wave = tid >> 5;
  const float thr = thr_p[0];

  const float* xrow    = x   + (size_t)row * kN;
  float*       featRow = out + (size_t)row * kN;
  float*       idxRow  = out + kFeatElems + (size_t)row * kK;
  float*       maskRow = out + kFeatElems + kIdxElems + (size_t)row * kN;

  // ---------- Phase A: async DMA the row from global into LDS ----------
#if defined(__HIP_DEVICE_COMPILE__) && __has_builtin(__builtin_amdgcn_global_load_async_to_lds_b128)
  #pragma unroll
  for (int i = 0; i < kVPT / 4; ++i) {
    const int col = (i << 10) + (tid << 2);           // coalesced 16B per lane
    __builtin_amdgcn_global_load_async_to_lds_b128(
        (AS1 v4i*)(xrow + col), (AS3 v4i*)(buf + col), 0, 0);
  }
  #if __has_builtin(__builtin_amdgcn_s_wait_asynccnt)
  __builtin_amdgcn_s_wait_asynccnt(0);
  #else
  asm volatile("s_wait_asynccnt 0" ::: "memory");
  #endif
#else
  #pragma unroll
  for (int i = 0; i < kVPT / 4; ++i) {
    const int col = (i << 10) + (tid << 2);
    *(float4*)&buf[col] = *(const float4*)&xrow[col];
  }
#endif

  // ---------- Phase B: threshold into regs, write mask, zero feat in LDS ----------
  // Each thread touches only its own LDS slots here, and those slots were
  // filled by this wave's own async loads => per-wave asynccnt wait suffices.
  float val[kVPT];
  #pragma unroll
  for (int i = 0; i < kVPT / 4; ++i) {
    const int col = (i << 10) + (tid << 2);
    float4 v = *(float4*)&buf[col];
    float4 m, z4;
    m.x = (v.x >= thr) ? 1.0f : 0.0f;
    m.y = (v.y >= thr) ? 1.0f : 0.0f;
    m.z = (v.z >= thr) ? 1.0f : 0.0f;
    m.w = (v.w >= thr) ? 1.0f : 0.0f;
    val[i * 4 + 0] = (v.x >= thr) ? v.x : 0.0f;
    val[i * 4 + 1] = (v.y >= thr) ? v.y : 0.0f;
    val[i * 4 + 2] = (v.z >= thr) ? v.z : 0.0f;
    val[i * 4 + 3] = (v.w >= thr) ? v.w : 0.0f;
    *(float4*)&maskRow[col] = m;            // valid = !(x < thr), streamed out
    z4.x = z4.y = z4.z = z4.w = 0.0f;
    *(float4*)&buf[col] = z4;               // buf becomes the feat row
  }

  // ---------- Phase C: iterative top-64 (descending, low-index tie-break) ----------
  const float NEG = -__builtin_inff();
  float lv; int li;
  // local argmax over this thread's 16 register-resident values
  auto localmax = [&]() {
    lv = NEG; li = 0x7FFFFFFF;
    #pragma unroll
    for (int j = 0; j < kVPT; ++j) {
      const int col = ((j >> 2) << 10) + (tid << 2) + (j & 3);
      const float v = val[j];
      const bool better = (v > lv) || (v == lv && col < li);
      lv = better ? v : lv;
      li = better ? col : li;
    }
  };
  localmax();

  for (int k = 0; k < kK; ++k) {
    float v = lv; int idx = li;
    #pragma unroll
    for (int off = 16; off >= 1; off >>= 1) {   // wave32 lexicographic max-reduce
      const float ov = __shfl_xor(v, off, 32);
      const int   oi = __shfl_xor(idx, off, 32);
      if (ov > v || (ov == v && oi < idx)) { v = ov; idx = oi; }
    }
    if (lane == 0) { redV[wave] = v; redI[wave] = idx; }
    __syncthreads();
    if (tid == 0) {
      float bv = redV[0]; int bi = redI[0];
      #pragma unroll
      for (int w = 1; w < kTPB / 32; ++w) {
        const float wv = redV[w]; const int wi = redI[w];
        if (wv > bv || (wv == bv && wi < bi)) { bv = wv; bi = wi; }
      }
      sBestI = bi;
      idxRow[k] = (float)bi;                 // indices emitted in descending order
      buf[bi] = 1.0f;                        // scatter into the feat row (LDS)
    }
    __syncthreads();
    const int bi = sBestI;
    if (tid == ((bi & 1023) >> 2)) {         // owning thread only
      #pragma unroll
      for (int j = 0; j < kVPT; ++j) {
        const int col = ((j >> 2) << 10) + (tid << 2) + (j & 3);
        if (col == bi) val[j] = NEG;         // remove winner
      }
      localmax();                            // refresh local candidate
    }
  }

  // ---------- Phase D: async store the composed feat row LDS -> global ----------
#if defined(__HIP_DEVICE_COMPILE__) && __has_builtin(__builtin_amdgcn_global_store_async_from_lds_b128)
  #pragma unroll
  for (int i = 0; i < kVPT / 4; ++i) {
    const int col = (i << 10) + (tid << 2);
    __builtin_amdgcn_global_store_async_from_lds_b128(
        (AS1 v4i*)(featRow + col), (AS3 v4i*)(buf + col), 0, 0);
  }
  #if __has_builtin(__builtin_amdgcn_s_wait_asynccnt)
  __builtin_amdgcn_s_wait_asynccnt(0);
  #else
  asm volatile("s_wait_asynccnt 0" ::: "memory");
  #endif
#else
  #pragma unroll
  for (int i = 0; i < kVPT / 4; ++i) {
    const int col = (i << 10) + (tid << 2);
    *(float4*)&featRow[col] = *(float4*)&buf[col];
  }
#endif
}

extern "C" void kernel_launch(void* const* d_in, const int* in_sizes, int n_in,
                              void* d_out, int out_size, void* d_ws, size_t ws_size,
                              hipStream_t stream) {
  (void)in_sizes; (void)n_in; (void)out_size; (void)d_ws; (void)ws_size;
  const float* x   = (const float*)d_in[0];
  const float* thr = (const float*)d_in[1];
  float* out = (float*)d_out;
  topk_scatter_kernel<<<dim3(kRows), dim3(kTPB), 0, stream>>>(x, thr, out);
}